// DeformedAgent_40424232190497
// MI455X (gfx1250) — compile-verified
//
#include <hip/hip_runtime.h>
#include <hip/hip_bf16.h>
#include <cmath>

// ---------------- problem constants (match reference) ----------------
#define B_      4096
#define SEQIN   61     // rows in input (row 0 = CLS, dropped)
#define L_      60     // seq after CLS drop
#define C_      512
#define KS_     9
#define AGENT_  5
#define STRIDE_ 12
#define LK_     5      // output agent positions
#define MROWS   (B_ * LK_)        // 20480 GEMM rows
#define MTILES  (MROWS / 16)      // 1280

typedef __attribute__((ext_vector_type(2))) float v2f;
typedef __attribute__((ext_vector_type(8))) float v8f;
typedef __attribute__((ext_vector_type(4))) unsigned int u32x4;
typedef __attribute__((ext_vector_type(8))) int i32x8;
typedef __attribute__((ext_vector_type(4))) int i32x4;

#if __has_builtin(__builtin_amdgcn_tensor_load_to_lds) && __has_builtin(__builtin_amdgcn_s_wait_tensorcnt)
#define USE_TDM 1
#else
#define USE_TDM 0
#endif

// =====================================================================
// Kernel 1: per-batch pipeline  conv(dw,9) -> LN(C) -> gelu -> 5-tap dot
//           -> tanh -> idx[b][lk]   (one workgroup per batch)
// q tiles are staged global->LDS with the Tensor Data Mover, double
// buffered so the DMA of block lk+1 overlaps compute of block lk.
// =====================================================================
__global__ __launch_bounds__(256)
void k_agent_idx(const float* __restrict__ q,      // (B,61,512)
                 const float* __restrict__ wdw,    // (512,1,9)
                 const float* __restrict__ bdw,    // (512)
                 const float* __restrict__ lnw,    // (512)
                 const float* __restrict__ lnb,    // (512)
                 const float* __restrict__ wpt,    // (1,512,5)
                 int*         __restrict__ idxbuf) // (B,5)
{
    const int b   = blockIdx.x;
    const int tid = threadIdx.x;

    __shared__ float sq[2][13 * 512];  // double-buffered staged conv inputs
    __shared__ float xb[5 * 512];      // conv outputs for 5 positions
    __shared__ float smu[5], srs[5];
    __shared__ float wsum[8];

    const int c0 = tid;
    const int c1 = tid + 256;

    // per-thread weights for its two channels
    float wd0[KS_], wd1[KS_];
#pragma unroll
    for (int k = 0; k < KS_; ++k) { wd0[k] = wdw[c0 * KS_ + k]; wd1[k] = wdw[c1 * KS_ + k]; }
    const float bd0 = bdw[c0], bd1 = bdw[c1];
    const float lw0 = lnw[c0], lw1 = lnw[c1];
    const float lb0 = lnb[c0], lb1 = lnb[c1];
    float wp0[AGENT_], wp1[AGENT_];
#pragma unroll
    for (int t = 0; t < AGENT_; ++t) { wp0[t] = wpt[c0 * AGENT_ + t]; wp1[t] = wpt[c1 * AGENT_ + t]; }

    // ref = ((int)linspace(1,365,5) - 1) * 59/364  (exact in fp32)
    const float refs[LK_] = {0.0f, 14.75f, 29.5f, 44.25f, 59.0f};

    const size_t qb = (size_t)b * SEQIN * C_;

    const int wv = tid >> 5;   // wave id (8 waves of 32)
    const int ln = tid & 31;   // lane in wave (wave32)

#if USE_TDM
    // Issue one TDM descriptor copying the q rows for agent block `lk`
    // into LDS buffer (lk & 1).  1-D tile, 4-byte elements, contiguous.
    auto tdm_issue = [&](int lk) {
        const int base = lk * STRIDE_;
        const int l0   = (lk == 0) ? 0 : base - 4;        // first seq row (CLS-dropped)
        const int rows = (lk == 0) ? 9 : 13;              // rows to copy
        const unsigned n = (unsigned)(rows * C_);         // elements (<= 6656)

        unsigned ldsoff = (unsigned)(uintptr_t)&sq[lk & 1][(lk == 0) ? 4 * C_ : 0];
        unsigned long long ga =
            (unsigned long long)(uintptr_t)(q + qb + (size_t)(l0 + 1) * C_);

        u32x4 g0;
        g0[0] = 1u;                                        // count=1, no gather/iterate
        g0[1] = ldsoff;                                    // lds_addr
        g0[2] = (unsigned)ga;                              // global_addr[31:0]
        g0[3] = (unsigned)((ga >> 32) & 0x1FFFFFFu)        // global_addr[56:32]
              | (2u << 30);                                // type = 2 (image)
        i32x8 g1;
        g1[0] = 0x00020000;                                // data_size = 4B
        g1[1] = (int)((n & 0xFFFFu) << 16);                // tensor_dim0[15:0]
        g1[2] = (int)((n >> 16) | (1u << 16));             // tensor_dim0[31:16] | tensor_dim1=1
        g1[3] = (int)((n & 0xFFFFu) << 16);                // tile_dim0 = n
        g1[4] = 0;                                         // tile_dim1/2 unused
        g1[5] = (int)n;                                    // tensor_dim0_stride
        g1[6] = 0;
        g1[7] = 0;
        const i32x4 gz4 = {0, 0, 0, 0};
        const i32x8 gz8 = {0, 0, 0, 0, 0, 0, 0, 0};
        __builtin_amdgcn_tensor_load_to_lds(g0, g1, gz4, gz4, gz8, 0);
    };

    // zero-fill the 4 left-pad rows of buffer 0 (lk==0 conv padding)
    for (int e = tid; e < 4 * C_; e += 256) sq[0][e] = 0.0f;
    if (tid < 32) { tdm_issue(0); tdm_issue(1); }
#endif

    for (int lk = 0; lk < LK_; ++lk) {
        const int base = lk * STRIDE_;
        const float* s = sq[lk & 1];

#if USE_TDM
        if (tid < 32) {
            if (lk == LK_ - 1) __builtin_amdgcn_s_wait_tensorcnt(0);
            else               __builtin_amdgcn_s_wait_tensorcnt(1);
        }
#else
        // fallback: plain cooperative staging
        for (int e = tid; e < 13 * 512; e += 256) {
            const int rl = e >> 9;
            const int c  = e & 511;
            const int l  = base - 4 + rl;                  // l <= 56 always
            sq[lk & 1][e] = (l >= 0) ? q[qb + (size_t)(l + 1) * C_ + c] : 0.0f;
        }
#endif
        __syncthreads();

        // ---- depthwise conv for the 5 tap positions of this block ----
#pragma unroll
        for (int t = 0; t < AGENT_; ++t) {
            float a0 = bd0, a1 = bd1;
#pragma unroll
            for (int k = 0; k < KS_; ++k) {
                const int row = (t + k) * 512;
                a0 += s[row + c0] * wd0[k];
                a1 += s[row + c1] * wd1[k];
            }
            xb[t * 512 + c0] = a0;
            xb[t * 512 + c1] = a1;
        }
        __syncthreads();

        // ---- LayerNorm stats: wave t reduces position t over C ----
        if (wv < AGENT_) {
            float s1 = 0.0f, s2 = 0.0f;
            for (int c = ln; c < C_; c += 32) {
                const float x = xb[wv * 512 + c];
                s1 += x; s2 += x * x;
            }
#pragma unroll
            for (int o = 16; o >= 1; o >>= 1) {
                s1 += __shfl_xor(s1, o, 32);
                s2 += __shfl_xor(s2, o, 32);
            }
            if (ln == 0) {
                const float mu  = s1 * (1.0f / 512.0f);
                const float var = s2 * (1.0f / 512.0f) - mu * mu;
                smu[wv] = mu;
                srs[wv] = rsqrtf(var + 1e-5f);
            }
        }
        __syncthreads();

        // ---- gelu(exact) + dot with w_pt over (t, c) ----
        float part = 0.0f;
#pragma unroll
        for (int t = 0; t < AGENT_; ++t) {
            const float mu = smu[t], rs = srs[t];
            float x0 = (xb[t * 512 + c0] - mu) * rs * lw0 + lb0;
            float x1 = (xb[t * 512 + c1] - mu) * rs * lw1 + lb1;
            x0 = 0.5f * x0 * (1.0f + erff(x0 * 0.70710678118654752f));
            x1 = 0.5f * x1 * (1.0f + erff(x1 * 0.70710678118654752f));
            part += x0 * wp0[t] + x1 * wp1[t];
        }
#pragma unroll
        for (int o = 16; o >= 1; o >>= 1) part += __shfl_xor(part, o, 32);
        if (ln == 0) wsum[wv] = part;
        __syncthreads();

        if (tid == 0) {
            float tot = 0.0f;
#pragma unroll
            for (int w = 0; w < 8; ++w) tot += wsum[w];
            const float off = tanhf(tot);
            const float pos = off + refs[lk];
            int iv = (int)pos;                    // trunc toward zero (== astype(int32))
            iv = iv < 0 ? 0 : (iv > L_ - 1 ? L_ - 1 : iv);
            idxbuf[b * LK_ + lk] = iv;
        }
        __syncthreads();   // all reads of sq[lk&1] / xb done

#if USE_TDM
        // prefetch block lk+2 into the buffer we just finished reading
        if (tid < 32 && lk + 2 < LK_) tdm_issue(lk + 2);
#endif
    }
}

// =====================================================================
// Kernel 2: gather X tile (reshape-semantics) + fp32 WMMA GEMM
//           Y[r, n] = sum_k X[r,k] * Wq[n,k] + bq[n]
//           one workgroup = 16 rows x 512 cols, K = 512
// =====================================================================
__global__ __launch_bounds__(256)
void k_gather_gemm(const float* __restrict__ tokens, // (B,61,512)
                   const float* __restrict__ wq,     // (512,512)
                   const float* __restrict__ bq,     // (512)
                   const int*   __restrict__ idxbuf, // (B,5)
                   float*       __restrict__ out)    // flat (B,5,512)
{
    const int tid = threadIdx.x;
    const int r0  = blockIdx.x * 16;        // first GEMM row of this tile

    __shared__ float XA[16 * 516];          // padded stride: 516 % 64 = 4 -> conflict-free

    // ---- gather A tile: X[r, c] = tokens_flat[b, i*60 + idx[b, p%5]],
    //      p = lk*512 + c, i = p/5   (reference's reshape gather) ----
    for (int e = tid; e < 16 * 512; e += 256) {
        const int mrow = e >> 9;
        const int c    = e & 511;
        const int r    = r0 + mrow;
        const int b    = r / 5;
        const int lk   = r - b * 5;
        const int p    = lk * 512 + c;
        const int i    = p / 5;
        const int j    = p - i * 5;
        const int m    = i * L_ + idxbuf[b * LK_ + j];   // < 60*512
        const int lo   = m >> 9;
        const int co   = m & 511;
        XA[mrow * 516 + c] = tokens[(size_t)b * SEQIN * C_ + (size_t)(lo + 1) * C_ + co];
    }
    __syncthreads();

    // ---- WMMA fp32 GEMM: 8 waves, each wave owns 64 N columns (4 tiles) ----
    const int wave = tid >> 5;
    const int lane = tid & 31;
    const int lm   = lane & 15;              // M (for A) / N (for B,D) index within tile
    const int hk   = (lane >> 4) << 1;       // lanes 0-15: K-pair {k,k+1}; 16-31: {k+2,k+3}
    const int n0   = wave * 64;

    v8f acc[4];
#pragma unroll
    for (int nt = 0; nt < 4; ++nt) acc[nt] = (v8f)0.0f;

    const float* arow = &XA[lm * 516];

    for (int k0 = 0; k0 < C_; k0 += 4) {
        const int kk = k0 + hk;
        const v2f a = *(const v2f*)(arow + kk);          // A 16x4 fragment (ISA layout)
#pragma unroll
        for (int nt = 0; nt < 4; ++nt) {
            const int n = n0 + nt * 16 + lm;
            const v2f bf = *(const v2f*)(wq + (size_t)n * C_ + kk);  // B[k][n] = Wq[n][k]
            acc[nt] = __builtin_amdgcn_wmma_f32_16x16x4_f32(
                false, a, false, bf, (short)0, acc[nt], false, false);
        }
    }

    // ---- bias + store (C/D layout: VGPR g -> M = g + 8*(lane>=16), N = lane&15) ----
    const int mtop = (lane >> 4) << 3;
#pragma unroll
    for (int nt = 0; nt < 4; ++nt) {
        const int n = n0 + nt * 16 + lm;
        const float bias = bq[n];
#pragma unroll
        for (int g = 0; g < 8; ++g) {
            const int mr = mtop + g;
            out[(size_t)(r0 + mr) * C_ + n] = acc[nt][g] + bias;
        }
    }
}

// =====================================================================
// Host launcher
// =====================================================================
extern "C" void kernel_launch(void* const* d_in, const int* in_sizes, int n_in,
                              void* d_out, int out_size, void* d_ws, size_t ws_size,
                              hipStream_t stream)
{
    const float* q      = (const float*)d_in[0];
    const float* tokens = (const float*)d_in[1];
    const float* w_dw   = (const float*)d_in[2];
    const float* b_dw   = (const float*)d_in[3];
    const float* ln_w   = (const float*)d_in[4];
    const float* ln_b   = (const float*)d_in[5];
    const float* w_pt   = (const float*)d_in[6];
    const float* w_q    = (const float*)d_in[7];
    const float* b_q    = (const float*)d_in[8];

    int* idxbuf = (int*)d_ws;   // B*5 ints

    k_agent_idx<<<B_, 256, 0, stream>>>(q, w_dw, b_dw, ln_w, ln_b, w_pt, idxbuf);
    k_gather_gemm<<<MTILES, 256, 0, stream>>>(tokens, w_q, b_q, idxbuf, (float*)d_out);
}